// VQVAELayer_1717986918622
// MI455X (gfx1250) — compile-verified
//
#include <hip/hip_runtime.h>

typedef __attribute__((ext_vector_type(2))) float v2f;
typedef __attribute__((ext_vector_type(8))) float v8f;

#define D_DIM       64
#define K_CODES     512
#define NCHUNK      32          // 512 / 16 codebook column chunks
#define KCHUNK      16          // 64 / 4 k-slices per WMMA accumulation
#define PACK_ELEMS  (NCHUNK * KCHUNK * 32)   // 16384 float2 = 128 KB

// ---------------------------------------------------------------------------
// Pre-pass: repack w [64, 512] into V_WMMA_F32_16X16X4_F32 B-fragment order
// plus per-column squared norms, into d_ws.
//   B 4x16 layout: VGPR0 = {K=0 lanes0-15, K=2 lanes16-31}, VGPR1 = {K=1, K=3}
//   entry(n, kc, lane) = float2( w[kbase][col], w[kbase+1][col] )
//     kbase = 4*kc + (lane<16 ? 0 : 2),  col = 16*n + (lane&15)
// ---------------------------------------------------------------------------
__global__ __launch_bounds__(256) void vq_pack_w(const float* __restrict__ w,
                                                 v2f* __restrict__ wpack,
                                                 float* __restrict__ wnorm) {
    int e = blockIdx.x * blockDim.x + threadIdx.x;
    if (e < PACK_ELEMS) {
        int lane  = e & 31;
        int idx2  = e >> 5;
        int kc    = idx2 & (KCHUNK - 1);
        int n     = idx2 >> 4;
        int kbase = 4 * kc + ((lane < 16) ? 0 : 2);
        int col   = 16 * n + (lane & 15);
        v2f p;
        p.x = w[kbase * K_CODES + col];
        p.y = w[(kbase + 1) * K_CODES + col];
        wpack[e] = p;
    }
    if (e < K_CODES) {
        float s = 0.f;
        #pragma unroll
        for (int k = 0; k < D_DIM; ++k) {
            float v = w[k * K_CODES + e];
            s += v * v;
        }
        wnorm[e] = s;
    }
}

// ---------------------------------------------------------------------------
// Main kernel: one wave handles TWO 16-row tiles of x per iteration so every
// B fragment / wnorm value feeds two independent WMMA accumulation chains.
//   scores[16x512] = -2 * (x_tile @ w) + ||e||^2   via V_WMMA_F32_16X16X4_F32
//   argmin per row (first-index tie-break), gather w column -> out.
// ---------------------------------------------------------------------------
template <bool PACKED>
__global__ __launch_bounds__(256) void vq_nearest(const float* __restrict__ x,
                                                  const float* __restrict__ w,
                                                  const v2f*   __restrict__ wpack,
                                                  const float* __restrict__ wnorm,
                                                  float* __restrict__ out,
                                                  int ntiles) {
    __shared__ float s_wn[K_CODES];   // used by fallback path only
    if (!PACKED) {
        for (int c = threadIdx.x; c < K_CODES; c += 256) {
            float s = 0.f;
            for (int k = 0; k < D_DIM; ++k) {
                float v = w[k * K_CODES + c];
                s += v * v;
            }
            s_wn[c] = s;
        }
        __syncthreads();
    }

    const int lane   = threadIdx.x & 31;
    const int wave   = (blockIdx.x * blockDim.x + threadIdx.x) >> 5;
    const int nwaves = (gridDim.x * blockDim.x) >> 5;
    const int half   = lane >> 4;        // 0: rows M=0..7, 1: rows M=8..15
    const int l16    = lane & 15;
    const int sel    = half * 2;         // A/B k-slice select within fragment

    const int npairs = (ntiles + 1) >> 1;

    for (int p = wave; p < npairs; p += nwaves) {
        const int t0 = p * 2;
        int       t1 = t0 + 1;
        if (t1 >= ntiles) t1 = ntiles - 1;   // clamp (no divergence; dup write ok)

        // ---- load A fragments for both tiles (reused for all 32 chunks)
        const float* xrow0 = x + (size_t)(t0 * 16 + l16) * D_DIM;
        const float* xrow1 = x + (size_t)(t1 * 16 + l16) * D_DIM;
        v2f a0[KCHUNK], a1[KCHUNK];
        #pragma unroll
        for (int kc = 0; kc < KCHUNK; ++kc) {
            a0[kc] = *(const v2f*)(xrow0 + 4 * kc + sel);
            a1[kc] = *(const v2f*)(xrow1 + 4 * kc + sel);
        }

        float    bv0[8], bv1[8];
        unsigned bi0[8], bi1[8];
        #pragma unroll
        for (int v = 0; v < 8; ++v) {
            bv0[v] = 3.4e38f; bi0[v] = 0u;
            bv1[v] = 3.4e38f; bi1[v] = 0u;
        }

        for (int n = 0; n < NCHUNK; ++n) {
            v8f acc0 = {};
            v8f acc1 = {};
            if (PACKED) {
                const v2f* bp = wpack + (size_t)n * KCHUNK * 32 + lane;
                #pragma unroll
                for (int kc = 0; kc < KCHUNK; ++kc) {
                    v2f b = bp[kc * 32];   // coalesced global_load_b64, L2-hot
                    acc0 = __builtin_amdgcn_wmma_f32_16x16x4_f32(
                        false, a0[kc], false, b, (short)0, acc0, false, false);
                    acc1 = __builtin_amdgcn_wmma_f32_16x16x4_f32(
                        false, a1[kc], false, b, (short)0, acc1, false, false);
                }
            } else {
                const int col = n * 16 + l16;
                #pragma unroll
                for (int kc = 0; kc < KCHUNK; ++kc) {
                    const int kbase = 4 * kc + sel;
                    v2f b;
                    b.x = w[kbase * K_CODES + col];
                    b.y = w[(kbase + 1) * K_CODES + col];
                    acc0 = __builtin_amdgcn_wmma_f32_16x16x4_f32(
                        false, a0[kc], false, b, (short)0, acc0, false, false);
                    acc1 = __builtin_amdgcn_wmma_f32_16x16x4_f32(
                        false, a1[kc], false, b, (short)0, acc1, false, false);
                }
            }
            // score = ||e||^2 - 2*dot ; column of this lane = 16n + l16
            const unsigned colg = (unsigned)(n * 16 + l16);
            const float wn = PACKED ? wnorm[colg] : s_wn[colg];
            #pragma unroll
            for (int v = 0; v < 8; ++v) {
                float s0 = __builtin_fmaf(-2.0f, acc0[v], wn);
                float s1 = __builtin_fmaf(-2.0f, acc1[v], wn);
                if (s0 < bv0[v]) { bv0[v] = s0; bi0[v] = colg; }
                if (s1 < bv1[v]) { bv1[v] = s1; bi1[v] = colg; }
            }
        }

        // ---- argmin reduction across the 16 lanes of each half-wave
        #pragma unroll
        for (int v = 0; v < 8; ++v) {
            float    c0 = bv0[v], c1 = bv1[v];
            unsigned d0 = bi0[v], d1 = bi1[v];
            #pragma unroll
            for (int m = 8; m >= 1; m >>= 1) {
                float    o0 = __shfl_xor(c0, m, 32);
                unsigned e0 = (unsigned)__shfl_xor((int)d0, m, 32);
                float    o1 = __shfl_xor(c1, m, 32);
                unsigned e1 = (unsigned)__shfl_xor((int)d1, m, 32);
                if (o0 < c0 || (o0 == c0 && e0 < d0)) { c0 = o0; d0 = e0; }
                if (o1 < c1 || (o1 == c1 && e1 < d1)) { c1 = o1; d1 = e1; }
            }
            bi0[v] = d0; bi1[v] = d1;
        }

        // ---- gather + store: half h writes rows v + 8*h; lane l16 writes
        //      floats [4*l16 .. 4*l16+3] of the 64-float output row.
        #pragma unroll
        for (int v = 0; v < 8; ++v) {
            const int      row0 = t0 * 16 + v + 8 * half;
            const int      row1 = t1 * 16 + v + 8 * half;
            const unsigned i0 = bi0[v];
            const unsigned i1 = bi1[v];
            float4 o0, o1;
            o0.x = w[(4 * l16 + 0) * K_CODES + i0];
            o0.y = w[(4 * l16 + 1) * K_CODES + i0];
            o0.z = w[(4 * l16 + 2) * K_CODES + i0];
            o0.w = w[(4 * l16 + 3) * K_CODES + i0];
            o1.x = w[(4 * l16 + 0) * K_CODES + i1];
            o1.y = w[(4 * l16 + 1) * K_CODES + i1];
            o1.z = w[(4 * l16 + 2) * K_CODES + i1];
            o1.w = w[(4 * l16 + 3) * K_CODES + i1];
            *(float4*)(out + (size_t)row0 * D_DIM + 4 * l16) = o0;
            *(float4*)(out + (size_t)row1 * D_DIM + 4 * l16) = o1;
        }
    }
}

// ---------------------------------------------------------------------------
extern "C" void kernel_launch(void* const* d_in, const int* in_sizes, int n_in,
                              void* d_out, int out_size, void* d_ws, size_t ws_size,
                              hipStream_t stream) {
    const float* x = (const float*)d_in[0];   // [512*512, 64] f32
    const float* w = (const float*)d_in[1];   // [64, 512] f32
    float* out = (float*)d_out;

    const int nrows  = in_sizes[0] / D_DIM;   // 262144
    const int ntiles = nrows / 16;            // 16384

    const size_t need = PACK_ELEMS * sizeof(v2f) + K_CODES * sizeof(float);
    if (ws_size >= need) {
        v2f*   wpack = (v2f*)d_ws;
        float* wnorm = (float*)((char*)d_ws + PACK_ELEMS * sizeof(v2f));
        vq_pack_w<<<(PACK_ELEMS + 255) / 256, 256, 0, stream>>>(w, wpack, wnorm);
        vq_nearest<true><<<1024, 256, 0, stream>>>(x, w, wpack, wnorm, out, ntiles);
    } else {
        vq_nearest<false><<<1024, 256, 0, stream>>>(x, w, nullptr, nullptr, out, ntiles);
    }
}